// EPG_58832462021100
// MI455X (gfx1250) — compile-verified
//
#include <hip/hip_runtime.h>
#include <math.h>

// MI455X / gfx1250, wave32. EPG recurrence: per-k complex 3x3 rotation done as a
// constant real 6x6 via V_WMMA_F32_16X16X4_F32 (K split 4+2), 16 pixels per wave
// in the N columns. State lives in LDS (ping-pong); shift folded into store
// addressing, decay into store multiply, conj into sign flip.
// B operands fetched as row-pairs so the backend can use DS_LOAD_2ADDR_B32.

typedef __attribute__((ext_vector_type(2))) float v2f;
typedef __attribute__((ext_vector_type(8))) float v8f;

#define KST   17      // k-states (kmax=16)
#define NR    104     // 102 component rows + 2 permanent zero rows (102,103)
#define ROWP  17      // row pitch in dwords (16 px + 1 pad -> conflict-free)
#define WAVES 4       // waves per block; 4*16 = 64 pixels/block
#define NPULSE 32

// alpha = pi/4, phi = 0
#define C2f 0.8535533905932737f   // cos^2(pi/8)
#define S2f 0.1464466094067263f   // sin^2(pi/8)
#define SAf 0.7071067811865476f   // sin(pi/4)
#define CAf 0.7071067811865476f   // cos(pi/4)
#define HSf 0.3535533905932738f   // sin(pi/4)/2

// Real 6x6 expansion of complex A. Component order per k:
// 0:re F+, 1:re F-, 2:re Z, 3:im F+, 4:im F-, 5:im Z.
__device__ const float M6t[6][6] = {
    { C2f,  S2f,  0.f,  0.f,  0.f,  SAf},
    { S2f,  C2f,  0.f,  0.f,  0.f, -SAf},
    { 0.f,  0.f,  CAf,  HSf, -HSf,  0.f},
    { 0.f,  0.f, -SAf,  C2f,  S2f,  0.f},
    { 0.f,  0.f,  SAf,  S2f,  C2f,  0.f},
    {-HSf,  HSf,  0.f,  0.f,  0.f,  CAf},
};

__global__ __launch_bounds__(WAVES * 32) void epg_wmma_kernel(
    const float* __restrict__ qm,   // (3, NPIX): PD, T1raw, T2raw planes
    float* __restrict__ out,        // (NPIX, 32)
    int npix)
{
    __shared__ float Sbuf[2][WAVES][NR][ROWP];   // 56,576 B
    __shared__ float Ebuf[WAVES][16][NPULSE + 1];// 8,448 B (padded vs bank conflicts)

    const int lane = threadIdx.x & 31;           // wave32
    const int w    = threadIdx.x >> 5;
    const int n    = lane & 15;                  // pixel column within tile
    const bool hi  = lane >= 16;
    const int pixBase = blockIdx.x * (WAVES * 16) + w * 16;

    // ---- constant A operands (ISA 16x4 f32 layout: lanes0-15 K=0/1, lanes16-31 K=2/3)
    v2f a1, a2;
    a1.x = (n < 6) ? M6t[n][hi ? 2 : 0] : 0.f;
    a1.y = (n < 6) ? M6t[n][hi ? 3 : 1] : 0.f;
    a2.x = (n < 6 && !hi) ? M6t[n][4] : 0.f;     // K=4
    a2.y = (n < 6 && !hi) ? M6t[n][5] : 0.f;     // K=5 (K=6,7 zero via pad rows)

    // ---- per-pixel decay (valid for lanes 0-15; hi lanes compute harmless dup)
    const int p = pixBase + n;
    const float pd = qm[p];
    const float T1 = qm[npix + p] * 5000.0f + 0.01f;
    const float T2 = qm[2 * npix + p] * 1500.0f + 0.01f;
    const float E1 = expf(-15.0f / T1);
    const float E2 = expf(-15.0f / T2);
    const float bb = 1.0f - E1;

    // ---- init this wave's LDS region (both buffers; rows 102/103 stay zero forever)
    float* base0 = &Sbuf[0][w][0][0];
    float* base1 = &Sbuf[1][w][0][0];
    for (int i = lane; i < NR * ROWP; i += 32) { base0[i] = 0.f; base1[i] = 0.f; }
    if (!hi) Sbuf[0][w][2][n] = 1.0f;            // FF0: re Z_0 = 1
    // wave-private region + in-order DS ops: no block barrier needed

    // per-lane row selectors for the B-operand pairs (rows are consecutive)
    const int selB1 = hi ? 2 : 0;                // b1 rows: 6k+sel, 6k+sel+1

    for (int t = 0; t < NPULSE; ++t) {
        const int cur = t & 1, nxt = cur ^ 1;
#pragma unroll
        for (int k = 0; k < KST; ++k) {
            // B operand (assumed 4x16 layout mirroring A): lanes0-15 K=0/1, lanes16-31 K=2/3.
            // Each v2f = two consecutive rows at the same column -> DS_LOAD_2ADDR_B32.
            const float* p1 = &Sbuf[cur][w][6 * k + selB1][n];
            const float* p2 = &Sbuf[cur][w][hi ? 102 : 6 * k + 4][n];
            v2f b1, b2;
            b1.x = p1[0];      // row 6k+0 (lo) / 6k+2 (hi)
            b1.y = p1[ROWP];   // row 6k+1 (lo) / 6k+3 (hi)
            b2.x = p2[0];      // row 6k+4 (lo) / zero row 102 (hi)
            b2.y = p2[ROWP];   // row 6k+5 (lo) / zero row 103 (hi)

            v8f acc = {};
            acc = __builtin_amdgcn_wmma_f32_16x16x4_f32(false, a1, false, b1,
                                                        (short)0, acc, false, false);
            acc = __builtin_amdgcn_wmma_f32_16x16x4_f32(false, a2, false, b2,
                                                        (short)0, acc, false, false);

            if (!hi) {  // D rows 0-5 live in lanes 0-15 of acc[0..5]
                const float reFp = acc[0], reFm = acc[1], reZ = acc[2];
                const float imFp = acc[3], imFm = acc[4], imZ = acc[5];

                if (k == 0)  // echo = Fj[F+_0] before shift; |echo| * PD
                    Ebuf[w][n][t] =
                        __builtin_amdgcn_sqrtf(reFp * reFp + imFp * imFp) * pd;

                // G = Fj @ S^T, FF' = conj(E*G + b), folded into stores:
                if (k < 16) {            // F+_{k+1} <- F+_k
                    Sbuf[nxt][w][6 * (k + 1) + 0][n] =  E2 * reFp;
                    Sbuf[nxt][w][6 * (k + 1) + 3][n] = -E2 * imFp;
                }
                if (k >= 1) {            // F-_{k-1} <- F-_k
                    Sbuf[nxt][w][6 * (k - 1) + 1][n] =  E2 * reFm;
                    Sbuf[nxt][w][6 * (k - 1) + 4][n] = -E2 * imFm;
                }
                if (k == 1) {            // quirk: F+_0 <- F-_1 as well
                    Sbuf[nxt][w][0][n] =  E2 * reFm;
                    Sbuf[nxt][w][3][n] = -E2 * imFm;
                }
                // Z_k stays; b adds (1-E1) at k=0
                Sbuf[nxt][w][6 * k + 2][n] =  E1 * reZ + (k == 0 ? bb : 0.f);
                Sbuf[nxt][w][6 * k + 5][n] = -E1 * imZ;
                if (k == 16) {           // F-_16 <- 0
                    Sbuf[nxt][w][97][n]  = 0.f;
                    Sbuf[nxt][w][100][n] = 0.f;
                }
            }
        }
    }

    // coalesced epilogue: out[p][t], 32 lanes sweep t
    for (int i = lane; i < 16 * NPULSE; i += 32)
        out[(pixBase + (i >> 5)) * NPULSE + (i & 31)] = Ebuf[w][i >> 5][i & 31];
}

extern "C" void kernel_launch(void* const* d_in, const int* in_sizes, int n_in,
                              void* d_out, int out_size, void* d_ws, size_t ws_size,
                              hipStream_t stream) {
    (void)n_in; (void)d_ws; (void)ws_size; (void)out_size;
    const float* qm = (const float*)d_in[0];       // (1,3,NPIX,1) flat
    float* out = (float*)d_out;                    // (NPIX, 32) flat
    const int npix = in_sizes[0] / 3;              // 262144
    // n_pulses=32, TR=15 are fixed by the reference; baked into the kernel.
    const int blocks = npix / (WAVES * 16);        // 4096
    epg_wmma_kernel<<<blocks, WAVES * 32, 0, stream>>>(qm, out, npix);
}